// GCN_57758720196950
// MI455X (gfx1250) — compile-verified
//
#include <hip/hip_runtime.h>
#include <math.h>

#define N_NODES 40960
#define N_EDGES 655360
#define E_TOT   (N_EDGES + N_NODES)
#define HID     128
#define HEADS   4
#define CH      32
#define N_GRAPHS 16
#define NEG_SLOPE 0.2f

typedef float v2f __attribute__((ext_vector_type(2)));
typedef float v8f __attribute__((ext_vector_type(8)));

// -------- fp32 WMMA GEMM: out[nrows x 128] = A[nrows x 128] @ W[128 x 128] + bias --------
// One wave computes a 16x16 output tile via 32 chained V_WMMA_F32_16X16X4_F32.
__global__ void gat_gemm128_kernel(const float* __restrict__ A,
                                   const float* __restrict__ W,
                                   const float* __restrict__ bias,
                                   float* __restrict__ out) {
    int wave  = (blockIdx.x * blockDim.x + threadIdx.x) >> 5;
    int lane  = threadIdx.x & 31;
    const int tilesN = HID / 16;                 // 8
    int tileM = wave / tilesN;
    int tileN = wave - tileM * tilesN;
    int g  = lane >> 4;                          // half-wave group
    int mn = lane & 15;
    int row = tileM * 16 + mn;                   // A-frag: M = lane%16
    int col = tileN * 16 + mn;                   // B/C/D: N = lane%16
    const float* __restrict__ arow = A + (size_t)row * HID;

    v8f acc = {};
#pragma unroll
    for (int k0 = 0; k0 < HID; k0 += 4) {
        int k = k0 + 2 * g;                      // A frag VGPR v holds K = 2*(lane/16)+v
        v2f a; a.x = arow[k];           a.y = arow[k + 1];
        v2f b; b.x = W[k * HID + col];  b.y = W[(k + 1) * HID + col];
        acc = __builtin_amdgcn_wmma_f32_16x16x4_f32(
            /*neg_a=*/false, a, /*neg_b=*/false, b,
            /*c_mod=*/(short)0, acc, /*reuse_a=*/false, /*reuse_b=*/false);
    }
    float bv = bias[col];
#pragma unroll
    for (int v = 0; v < 8; ++v) {                // C/D: VGPR v -> M = v + 8*(lane/16)
        int orow = tileM * 16 + v + 8 * g;
        out[(size_t)orow * HID + col] = acc[v] + bv;
    }
}

// -------- per-layer state init --------
__global__ void gat_init_kernel(float* __restrict__ num,
                                float* __restrict__ emax,
                                float* __restrict__ denom) {
    int i = blockIdx.x * blockDim.x + threadIdx.x;
    if (i < N_NODES * HID) num[i] = 0.f;
    if (i < N_NODES * HEADS) { emax[i] = -INFINITY; denom[i] = 0.f; }
}

__device__ __forceinline__ void atomicMaxF(float* addr, float v) {
    // order-independent float atomic max via monotonic int/uint punning
    if (v >= 0.f) atomicMax((int*)addr, __float_as_int(v));
    else          atomicMin((unsigned int*)addr, __float_as_uint(v));
}

// -------- edge pass 1: logits e[E,4] + segment max over dst --------
__global__ void gat_edge_att_kernel(const int* __restrict__ ei,
                                    const float* __restrict__ xl,
                                    const float* __restrict__ xr,
                                    const float* __restrict__ att,
                                    float* __restrict__ ebuf,
                                    float* __restrict__ emax) {
    int wid  = (blockIdx.x * blockDim.x + threadIdx.x) >> 5;   // one wave per edge
    int lane = threadIdx.x & 31;                               // channel within head
    if (wid >= E_TOT) return;
    int s, d;
    if (wid < N_EDGES) { s = ei[wid]; d = ei[N_EDGES + wid]; }
    else               { s = d = wid - N_EDGES; }              // self loop
    const float* __restrict__ xls = xl + (size_t)s * HID;
    const float* __restrict__ xrd = xr + (size_t)d * HID;

    float t[HEADS];
#pragma unroll
    for (int h = 0; h < HEADS; ++h) {
        float v  = xls[h * CH + lane] + xrd[h * CH + lane];
        float lr = v > 0.f ? v : NEG_SLOPE * v;
        t[h] = lr * att[h * CH + lane];
    }
#pragma unroll
    for (int h = 0; h < HEADS; ++h)
#pragma unroll
        for (int m = 16; m >= 1; m >>= 1)
            t[h] += __shfl_xor(t[h], m, 32);                   // wave32 butterfly

    if (lane < HEADS) {
        float ev = (lane == 0) ? t[0] : (lane == 1) ? t[1] : (lane == 2) ? t[2] : t[3];
        ebuf[(size_t)wid * HEADS + lane] = ev;
        atomicMaxF(&emax[d * HEADS + lane], ev);
    }
}

// -------- edge pass 2: ex = exp(e-max); denom += ex; num[dst] += ex * xl[src] --------
__global__ void gat_edge_agg_kernel(const int* __restrict__ ei,
                                    const float* __restrict__ xl,
                                    const float* __restrict__ ebuf,
                                    const float* __restrict__ emax,
                                    float* __restrict__ denom,
                                    float* __restrict__ num) {
    int wid  = (blockIdx.x * blockDim.x + threadIdx.x) >> 5;
    int lane = threadIdx.x & 31;
    if (wid >= E_TOT) return;
    int s, d;
    if (wid < N_EDGES) { s = ei[wid]; d = ei[N_EDGES + wid]; }
    else               { s = d = wid - N_EDGES; }

    float ex[HEADS];
#pragma unroll
    for (int h = 0; h < HEADS; ++h)
        ex[h] = expf(ebuf[(size_t)wid * HEADS + h] - emax[d * HEADS + h]);

    if (lane < HEADS) {
        float e0 = (lane == 0) ? ex[0] : (lane == 1) ? ex[1] : (lane == 2) ? ex[2] : ex[3];
        atomicAdd(&denom[d * HEADS + lane], e0);
    }
    const float* __restrict__ xls = xl + (size_t)s * HID;
    float* __restrict__ nd = num + (size_t)d * HID;
#pragma unroll
    for (int h = 0; h < HEADS; ++h)
        atomicAdd(&nd[h * CH + lane], ex[h] * xls[h * CH + lane]);
}

// -------- node finalize: h = num/denom + bias (+ ReLU) --------
__global__ void gat_finalize_kernel(const float* __restrict__ num,
                                    const float* __restrict__ denom,
                                    const float* __restrict__ bias,
                                    float* __restrict__ hout, int relu) {
    int i = blockIdx.x * blockDim.x + threadIdx.x;
    if (i >= N_NODES * HID) return;
    int n  = i >> 7;
    int ch = i & 127;
    int h  = ch >> 5;
    float v = num[i] / denom[n * HEADS + h] + bias[ch];
    if (relu) v = fmaxf(v, 0.f);
    hout[i] = v;
}

// -------- global mean pool --------
__global__ void pool_zero_kernel(float* __restrict__ sums, float* __restrict__ cnt) {
    int i = blockIdx.x * blockDim.x + threadIdx.x;
    if (i < N_GRAPHS * HID) sums[i] = 0.f;
    if (i < N_GRAPHS) cnt[i] = 0.f;
}

__global__ void pool_acc_kernel(const float* __restrict__ h,
                                const int* __restrict__ batch,
                                float* __restrict__ sums,
                                float* __restrict__ cnt) {
    int i = blockIdx.x * blockDim.x + threadIdx.x;
    if (i >= N_NODES * HID) return;
    int n  = i >> 7;
    int ch = i & 127;
    int g  = batch[n];
    atomicAdd(&sums[g * HID + ch], h[i]);
    if (ch == 0) atomicAdd(&cnt[g], 1.f);
}

__global__ void pool_fin_kernel(const float* __restrict__ sums,
                                const float* __restrict__ cnt,
                                float* __restrict__ pooled) {
    int i = blockIdx.x * blockDim.x + threadIdx.x;
    if (i >= N_GRAPHS * HID) return;
    pooled[i] = sums[i] / fmaxf(cnt[i >> 7], 1.f);
}

extern "C" void kernel_launch(void* const* d_in, const int* in_sizes, int n_in,
                              void* d_out, int out_size, void* d_ws, size_t ws_size,
                              hipStream_t stream) {
    (void)in_sizes; (void)n_in; (void)out_size; (void)ws_size;
    const float* x     = (const float*)d_in[0];
    const int*   ei    = (const int*)d_in[1];
    const int*   batch = (const int*)d_in[2];

    char* ws = (char*)d_ws;
    const size_t NB = (size_t)N_NODES * HID * sizeof(float);      // 20.97 MB
    const size_t NH = (size_t)N_NODES * HEADS * sizeof(float);    // 0.64 MB
    float* xl    = (float*)(ws);
    float* xr    = (float*)(ws + NB);
    float* num   = (float*)(ws + 2 * NB);
    float* bufH  = (float*)(ws + 3 * NB);
    float* emax  = (float*)(ws + 4 * NB);
    float* denom = (float*)(ws + 4 * NB + NH);
    float* ebuf  = (float*)(ws + 4 * NB + 2 * NH);
    float* sums  = (float*)(ws + 4 * NB + 2 * NH + (size_t)E_TOT * HEADS * sizeof(float));
    float* cnt   = sums + N_GRAPHS * HID;

    float* out_pooled = (float*)d_out;
    float* out_h      = out_pooled + N_GRAPHS * HID;

    const int gemmBlocks = (N_NODES / 16) * (HID / 16) / 8;       // 8 waves/block
    const int nodeBlocks = (N_NODES * HID + 255) / 256;
    const int edgeBlocks = (E_TOT + 7) / 8;                       // 8 edges(waves)/block

    const float* hin = x;
    for (int l = 0; l < 3; ++l) {
        const float* Wl = (const float*)d_in[3 + 6 * l + 0];
        const float* bl = (const float*)d_in[3 + 6 * l + 1];
        const float* Wr = (const float*)d_in[3 + 6 * l + 2];
        const float* br = (const float*)d_in[3 + 6 * l + 3];
        const float* at = (const float*)d_in[3 + 6 * l + 4];
        const float* bb = (const float*)d_in[3 + 6 * l + 5];

        gat_gemm128_kernel<<<gemmBlocks, 256, 0, stream>>>(hin, Wl, bl, xl);
        gat_gemm128_kernel<<<gemmBlocks, 256, 0, stream>>>(hin, Wr, br, xr);
        gat_init_kernel<<<nodeBlocks, 256, 0, stream>>>(num, emax, denom);
        gat_edge_att_kernel<<<edgeBlocks, 256, 0, stream>>>(ei, xl, xr, at, ebuf, emax);
        gat_edge_agg_kernel<<<edgeBlocks, 256, 0, stream>>>(ei, xl, ebuf, emax, denom, num);

        float* hout = (l == 2) ? out_h : bufH;
        gat_finalize_kernel<<<nodeBlocks, 256, 0, stream>>>(num, denom, bb, hout, l < 2 ? 1 : 0);
        hin = bufH;
    }

    pool_zero_kernel<<<(N_GRAPHS * HID + 255) / 256, 256, 0, stream>>>(sums, cnt);
    pool_acc_kernel<<<nodeBlocks, 256, 0, stream>>>(out_h, batch, sums, cnt);
    pool_fin_kernel<<<(N_GRAPHS * HID + 255) / 256, 256, 0, stream>>>(sums, cnt, out_pooled);
}